// SimpleGNN_12859132084712
// MI455X (gfx1250) — compile-verified
//
#include <hip/hip_runtime.h>

typedef __attribute__((ext_vector_type(2))) float v2f;
typedef __attribute__((ext_vector_type(8))) float v8f;

#define GCN_N 50000
#define GCN_E 800000
#define GCN_K 128   // DIN == H
#define GCN_C 40
#define GCN_CP 48   // C padded to a multiple of 16

// ---------------------------------------------------------------- utilities
__global__ void zero_f32(float* __restrict__ p, int count) {
    int i = blockIdx.x * blockDim.x + threadIdx.x;
    if (i < count) p[i] = 0.0f;
}

// Histogram both endpoint degrees with float atomics (counts are exact in f32).
__global__ void edge_degrees(const int* __restrict__ src, const int* __restrict__ dst,
                             float* __restrict__ deg_src, float* __restrict__ deg_dst, int E) {
    int e = blockIdx.x * blockDim.x + threadIdx.x;
    if (e >= E) return;
    atomicAdd(&deg_src[src[e]], 1.0f);
    atomicAdd(&deg_dst[dst[e]], 1.0f);
}

// deg -> (deg > 0 ? rsqrt(deg) : 1) in place (both arrays contiguous)
__global__ void finalize_norm(float* __restrict__ deg, int count) {
    int i = blockIdx.x * blockDim.x + threadIdx.x;
    if (i >= count) return;
    float d = deg[i];
    deg[i] = (d > 0.0f) ? rsqrtf(d) : 1.0f;
}

// Zero-pad Wfc (K x 40) -> Wp (K x 48), bfc (40) -> bp (48), so the FC GEMM
// runs with compile-time stride and no load guards.
__global__ void pad_fc(const float* __restrict__ Wfc, const float* __restrict__ bfc,
                       float* __restrict__ Wp, float* __restrict__ bp) {
    int i = blockIdx.x * blockDim.x + threadIdx.x;
    if (i < GCN_K * GCN_CP) {
        int r = i / GCN_CP, c = i - r * GCN_CP;
        Wp[i] = (c < GCN_C) ? Wfc[r * GCN_C + c] : 0.0f;
    }
    if (i < GCN_CP) bp[i] = (i < GCN_C) ? bfc[i] : 0.0f;
}

// ------------------------------------------------------- edge scatter (hot)
// One wave per edge; edge state scalarized via readfirstlane so src/dst/norm
// become scalar loads. Lane l moves features [4l,4l+4): one coalesced b128
// gather of the source row + 4x global_atomic_add_f32 into the contiguous
// 512B destination row.
__global__ void scatter_edges(const float* __restrict__ x,
                              const int* __restrict__ src, const int* __restrict__ dst,
                              const float* __restrict__ norm_src,
                              float* __restrict__ agg, int E) {
    int ew   = blockIdx.x * (blockDim.x >> 5) + (threadIdx.x >> 5);
    int e    = __builtin_amdgcn_readfirstlane(ew);   // wave-uniform -> SGPR
    int lane = threadIdx.x & 31;
    if (e >= E) return;
    int s = __builtin_amdgcn_readfirstlane(src[e]);
    int d = __builtin_amdgcn_readfirstlane(dst[e]);
    float ns = norm_src[s];
    const float4* xrow = (const float4*)(x + (size_t)s * GCN_K);
    float4 v = xrow[lane];
    float* arow = agg + (size_t)d * GCN_K + lane * 4;
    atomicAdd(arow + 0, v.x * ns);
    atomicAdd(arow + 1, v.y * ns);
    atomicAdd(arow + 2, v.z * ns);
    atomicAdd(arow + 3, v.w * ns);
}

// ------------------------------------------------------- fp32 WMMA GEMM
// out[n_rows x OUT_COLS] = act( (A .* norm) @ W + bias ), A is n_rows x K,
// W is K x NCOLS (compile-time stride => immediate-offset loads, no guards,
// fully unrolled: 32x v_wmma_f32_16x16x4_f32 per 16x16 tile). Only the final
// store masks columns when OUT_COLS < NCOLS; EXEC is all-ones through WMMA.
template <int NCOLS, int OUT_COLS, bool NORM, bool RELU>
__global__ void gemm_wmma_f32(const float* __restrict__ A,
                              const float* __restrict__ norm_dst,
                              const float* __restrict__ W,
                              const float* __restrict__ bias,
                              float* __restrict__ out, int n_rows) {
    constexpr int TILES_N = NCOLS / 16;
    int tile = blockIdx.x * (blockDim.x >> 5) + (threadIdx.x >> 5);
    int tiles_total = (n_rows >> 4) * TILES_N;
    if (tile >= tiles_total) return;                 // wave-uniform exit
    int tile_m = tile / TILES_N;
    int tile_n = tile - tile_m * TILES_N;

    int lane = threadIdx.x & 31;
    int half = lane >> 4;        // 0: lanes 0-15, 1: lanes 16-31
    int m16  = lane & 15;
    int kb   = half * 2;         // k-pair base per ISA f32 16x4 / 4x16 layout

    int row_a = tile_m * 16 + m16;
    int col_b = tile_n * 16 + m16;

    float scale = NORM ? norm_dst[row_a] : 1.0f;
    const float* arow = A + (size_t)row_a * GCN_K + kb;   // 8B-aligned (kb even)
    const float* wcol = W + (size_t)kb * NCOLS + col_b;

    v8f acc = {};
#pragma unroll
    for (int k = 0; k < GCN_K; k += 4) {
        v2f a = *(const v2f*)(arow + k);                  // global_load_b64 offset:imm
        a.x *= scale;
        a.y *= scale;                                     // v_pk_mul_f32
        v2f b;
        b.x = wcol[(size_t)k * NCOLS];                    // global_load_b32 offset:imm
        b.y = wcol[(size_t)(k + 1) * NCOLS];
        acc = __builtin_amdgcn_wmma_f32_16x16x4_f32(
                  false, a, false, b, (short)0, acc, false, false);
    }

    float bv = bias[col_b];
    // D layout: VGPR v holds M=v (lanes 0-15) and M=v+8 (lanes 16-31), N=lane%16.
#pragma unroll
    for (int v = 0; v < 8; ++v) {
        int m = v + half * 8;
        float val = acc[v] + bv;
        if (RELU) val = fmaxf(val, 0.0f);
        if (OUT_COLS == NCOLS || col_b < OUT_COLS)
            out[(size_t)(tile_m * 16 + m) * OUT_COLS + col_b] = val;
    }
}

// ---------------------------------------------------------------- driver
extern "C" void kernel_launch(void* const* d_in, const int* in_sizes, int n_in,
                              void* d_out, int out_size, void* d_ws, size_t ws_size,
                              hipStream_t stream) {
    const float* x   = (const float*)d_in[0];
    const int*   src = (const int*)  d_in[1];
    const int*   dst = (const int*)  d_in[2];
    const float* W1  = (const float*)d_in[3];
    const float* b1  = (const float*)d_in[4];
    const float* W2  = (const float*)d_in[5];
    const float* b2  = (const float*)d_in[6];
    const float* Wfc = (const float*)d_in[7];
    const float* bfc = (const float*)d_in[8];
    float* out = (float*)d_out;

    const int N = GCN_N, E = GCN_E, K = GCN_K;

    float* ws       = (float*)d_ws;
    float* norm_src = ws;                          // N
    float* norm_dst = ws + N;                      // N
    float* agg      = ws + 2 * N;                  // N*K
    float* h        = agg + (size_t)N * K;         // N*K
    float* wfc_pad  = h + (size_t)N * K;           // K*48
    float* bfc_pad  = wfc_pad + GCN_K * GCN_CP;    // 48

    const int TB = 256;
    auto blocks = [](long long n, int tb) { return (int)((n + tb - 1) / tb); };

    // 1) degree norms + FC weight padding
    zero_f32<<<blocks(2 * N, TB), TB, 0, stream>>>(norm_src, 2 * N);
    edge_degrees<<<blocks(E, TB), TB, 0, stream>>>(src, dst, norm_src, norm_dst, E);
    finalize_norm<<<blocks(2 * N, TB), TB, 0, stream>>>(norm_src, 2 * N);
    pad_fc<<<blocks(GCN_K * GCN_CP, TB), TB, 0, stream>>>(Wfc, bfc, wfc_pad, bfc_pad);

    const int scatter_blocks = blocks((long long)E * 32, TB);
    const int gemm_blocks_hh = blocks((N / 16) * (K / 16), TB / 32);       // 25000 tiles
    const int gemm_blocks_fc = blocks((N / 16) * (GCN_CP / 16), TB / 32);  // 9375 tiles

    // 2) layer 1: scatter(x * norm_src) -> agg; h = relu((agg*norm_dst)@W1+b1)
    zero_f32<<<blocks((long long)N * K, TB), TB, 0, stream>>>(agg, N * K);
    scatter_edges<<<scatter_blocks, TB, 0, stream>>>(x, src, dst, norm_src, agg, E);
    gemm_wmma_f32<GCN_K, GCN_K, true, true>
        <<<gemm_blocks_hh, TB, 0, stream>>>(agg, norm_dst, W1, b1, h, N);

    // 3) layer 2: scatter(h * norm_src) -> agg; h = relu((agg*norm_dst)@W2+b2)
    zero_f32<<<blocks((long long)N * K, TB), TB, 0, stream>>>(agg, N * K);
    scatter_edges<<<scatter_blocks, TB, 0, stream>>>(h, src, dst, norm_src, agg, E);
    gemm_wmma_f32<GCN_K, GCN_K, true, true>
        <<<gemm_blocks_hh, TB, 0, stream>>>(agg, norm_dst, W2, b2, h, N);

    // 4) FC head: out = h @ Wfc_pad + bfc_pad (store-masked to 40 cols)
    gemm_wmma_f32<GCN_CP, GCN_C, false, false>
        <<<gemm_blocks_fc, TB, 0, stream>>>(h, nullptr, wfc_pad, bfc_pad, out, N);
}